// RelationClassifier_75179107549655
// MI455X (gfx1250) — compile-verified
//
#include <hip/hip_runtime.h>
#include <hip/hip_bf16.h>
#include <stddef.h>

// Problem constants (match reference)
#define B_    4
#define L_    512
#define H_    768
#define R_    128
#define E_    64
#define SEMB  25
#define FEAT_ 2379
#define KPAD  2432          // FEAT padded to multiple of 4 (and 64)
#define KG    (KPAD / 4)    // 608 K-groups of 4
#define TILE_R 8

// feature-row layout (matches jnp.concatenate order)
#define OFF_SPAN0 0
#define OFF_SPAN1 768
#define OFF_SIZE0 1536
#define OFF_SIZE1 1561
#define OFF_CTX   1586
#define OFF_CSIZE 2354

// output layout: type[4,128,10] ++ cate[4,128,12] ++ pola[4,128,3]
#define OUT_CATE_OFF 5120
#define OUT_POLA_OFF 11264

typedef float v2f __attribute__((ext_vector_type(2)));
typedef float v8f __attribute__((ext_vector_type(8)));

// ---------------------------------------------------------------------------
// Kernel 1: build padded feature matrix feat[512][KPAD]
//   one block = (one batch b, 8 relations); hidden[b] read once per block,
//   masks packed to bitmasks in LDS, inner loop = coalesced loads + v_max_f32
// ---------------------------------------------------------------------------
__global__ __launch_bounds__(256) void rel_build_features(
    const float* __restrict__ hidden,      // [B,L,H]
    const int*   __restrict__ rel_pairs,   // [B,R,2]
    const int*   __restrict__ rel_masks,   // [B,R,L]
    const int*   __restrict__ rel_sizes,   // [B,R]
    const float* __restrict__ span_reps,   // [B,E,H]
    const float* __restrict__ size_embs,   // [B,E,SEMB]
    const float* __restrict__ size_table,  // [30,SEMB]
    float*       __restrict__ feat)        // [512,KPAD]
{
    __shared__ unsigned mbits[TILE_R * 16];   // 8 relations x 16 x 32-bit mask words

    const int blk = blockIdx.x;
    const int b   = blk / (R_ / TILE_R);
    const int r0  = (blk % (R_ / TILE_R)) * TILE_R;
    const int tid = threadIdx.x;

    // pack masks into bitmasks (128 words, one per thread)
    if (tid < TILE_R * 16) {
        const int r = tid >> 4;
        const int w = tid & 15;
        const int* mp = rel_masks + ((size_t)(b * R_ + r0 + r) * L_ + w * 32);
        unsigned bits = 0u;
        #pragma unroll
        for (int i = 0; i < 32; ++i)
            bits |= (mp[i] != 0) ? (1u << i) : 0u;
        mbits[tid] = bits;
    }
    __syncthreads();

    // masked max-pool: thread covers h = tid, tid+256, tid+512
    float acc[3][TILE_R];
    #pragma unroll
    for (int i = 0; i < 3; ++i)
        #pragma unroll
        for (int r = 0; r < TILE_R; ++r)
            acc[i][r] = -3.0e38f;

    const float* hb = hidden + (size_t)b * L_ * H_;
    for (int lw = 0; lw < 16; ++lw) {
        unsigned wb[TILE_R];
        unsigned anyw = 0u;
        #pragma unroll
        for (int r = 0; r < TILE_R; ++r) { wb[r] = mbits[r * 16 + lw]; anyw |= wb[r]; }
        if (!anyw) continue;                      // uniform: same LDS data blockwide
        #pragma unroll 4
        for (int lb = 0; lb < 32; ++lb) {
            const int l = lw * 32 + lb;
            const float* hp = hb + (size_t)l * H_ + tid;
            if (l + 1 < L_) __builtin_prefetch((const void*)(hp + H_), 0, 1);
            float hv0 = hp[0], hv1 = hp[256], hv2 = hp[512];
            #pragma unroll
            for (int r = 0; r < TILE_R; ++r) {
                if ((wb[r] >> lb) & 1u) {
                    acc[0][r] = fmaxf(acc[0][r], hv0);
                    acc[1][r] = fmaxf(acc[1][r], hv1);
                    acc[2][r] = fmaxf(acc[2][r], hv2);
                }
            }
        }
    }

    // write ctx + gathers + padding per relation
    for (int r = 0; r < TILE_R; ++r) {
        const int rr = r0 + r;
        float* frow = feat + (size_t)(b * R_ + rr) * KPAD;

        unsigned any = 0u;
        #pragma unroll
        for (int w = 0; w < 16; ++w) any |= mbits[r * 16 + w];

        #pragma unroll
        for (int i = 0; i < 3; ++i) {
            const int h = tid + i * 256;
            frow[OFF_CTX + h] = any ? acc[i][r] : 0.0f;
        }

        const int e0 = rel_pairs[(size_t)(b * R_ + rr) * 2 + 0];
        const int e1 = rel_pairs[(size_t)(b * R_ + rr) * 2 + 1];
        const float* s0 = span_reps + ((size_t)b * E_ + e0) * H_;
        const float* s1 = span_reps + ((size_t)b * E_ + e1) * H_;
        #pragma unroll
        for (int i = 0; i < 3; ++i) {
            const int h = tid + i * 256;
            frow[OFF_SPAN0 + h] = s0[h];
            frow[OFF_SPAN1 + h] = s1[h];
        }

        if (tid < SEMB) {
            const float* z0 = size_embs + ((size_t)b * E_ + e0) * SEMB;
            const float* z1 = size_embs + ((size_t)b * E_ + e1) * SEMB;
            const float* zt = size_table + (size_t)rel_sizes[b * R_ + rr] * SEMB;
            frow[OFF_SIZE0 + tid] = z0[tid];
            frow[OFF_SIZE1 + tid] = z1[tid];
            frow[OFF_CSIZE + tid] = zt[tid];
        }
        // zero-pad K tail [FEAT_, KPAD)
        for (int k = FEAT_ + tid; k < KPAD; k += 256) frow[k] = 0.0f;
    }
}

// ---------------------------------------------------------------------------
// Kernel 2: pack W_type/W_cate/W_pola (+ biases) into the exact per-lane
// B-operand layout of V_WMMA_F32_16X16X4_F32:
//   PB[ntile][g][lane][j] = W[n][k],  k = 4g + 2*(lane>=16) + j,  n = 16*ntile + (lane&15)
// so the GEMM's B fetch is one coalesced global_load_b64 per lane.
// ---------------------------------------------------------------------------
__global__ __launch_bounds__(256) void rel_pack_weights(
    const float* __restrict__ Wt, const float* __restrict__ bt,
    const float* __restrict__ Wc, const float* __restrict__ bc,
    const float* __restrict__ Wp, const float* __restrict__ bp,
    float* __restrict__ PB,     // [2][KG][64]
    float* __restrict__ bias)   // [32]
{
    const int t = blockIdx.x * blockDim.x + threadIdx.x;
    if (t < 32) {
        const int c = t;
        float bv = 0.0f;
        if (c < 10)      bv = bt[c];
        else if (c < 22) bv = bc[c - 10];
        else if (c < 25) bv = bp[c - 22];
        bias[c] = bv;
    }
    if (t >= 2 * KG * 64) return;
    const int ntile = t / (KG * 64);
    const int rem   = t % (KG * 64);
    const int g     = rem >> 6;
    const int idx   = rem & 63;
    const int lane  = idx >> 1;
    const int j     = idx & 1;
    const int k     = g * 4 + ((lane >> 4) << 1) + j;
    const int n     = ntile * 16 + (lane & 15);
    float v = 0.0f;
    if (k < FEAT_ && n < 25) {
        if (n < 10)      v = Wt[(size_t)n * FEAT_ + k];
        else if (n < 22) v = Wc[(size_t)(n - 10) * FEAT_ + k];
        else             v = Wp[(size_t)(n - 22) * FEAT_ + k];
    }
    PB[t] = v;
}

// ---------------------------------------------------------------------------
// Kernel 3: WMMA GEMM  out[512,25(+bias)] = feat[512,KPAD] @ Wt[KPAD,32]
//   4 waves / block, one 16-row M-tile per wave, two 16-col N-tiles per wave,
//   K-loop of KG=608 steps of V_WMMA_F32_16X16X4_F32. EXEC is all-ones.
// ---------------------------------------------------------------------------
__global__ __launch_bounds__(128) void rel_gemm_wmma(
    const float* __restrict__ feat,   // [512,KPAD]
    const float* __restrict__ PB,     // [2][KG][64]
    const float* __restrict__ bias,   // [32]
    float*       __restrict__ out)    // 12800 floats
{
    const int lane  = threadIdx.x & 31;
    const int wave  = threadIdx.x >> 5;
    const int mtile = blockIdx.x * 4 + wave;     // 0..31
    const int m0    = mtile * 16;
    const int hi    = lane >> 4;                 // 0: K pair {0,1}; 1: K pair {2,3}
    const int lm    = lane & 15;                 // M (for A) / N (for B,D)

    // A: lane holds feat[m0+lm][k0 + 2*hi .. +1]  -> one b64 load per step
    const float* arow = feat + (size_t)(m0 + lm) * KPAD + (hi << 1);
    const float* b0p  = PB + (size_t)lane * 2;                       // ntile 0
    const float* b1p  = PB + (size_t)KG * 64 + (size_t)lane * 2;     // ntile 1

    v8f acc0 = {0.f, 0.f, 0.f, 0.f, 0.f, 0.f, 0.f, 0.f};
    v8f acc1 = {0.f, 0.f, 0.f, 0.f, 0.f, 0.f, 0.f, 0.f};

    for (int g = 0; g < KG; ++g) {
        v2f a  = *(const v2f*)(arow + (size_t)g * 4);
        v2f b0 = *(const v2f*)(b0p  + (size_t)g * 64);
        v2f b1 = *(const v2f*)(b1p  + (size_t)g * 64);
        acc0 = __builtin_amdgcn_wmma_f32_16x16x4_f32(false, a, false, b0,
                                                     (short)0, acc0, false, false);
        acc1 = __builtin_amdgcn_wmma_f32_16x16x4_f32(false, a, false, b1,
                                                     (short)0, acc1, false, false);
    }

    // Epilogue: D layout (16x16 f32): VGPR v -> M = v + 8*hi, N = lm.
    #pragma unroll
    for (int nt = 0; nt < 2; ++nt) {
        const v8f acc = nt ? acc1 : acc0;
        const int c = nt * 16 + lm;              // global class column
        if (c >= 25) continue;
        const float bv = bias[c];
        #pragma unroll
        for (int v = 0; v < 8; ++v) {
            const int m   = v + (hi << 3);
            const int row = m0 + m;              // b*R + r
            const float val = acc[v] + bv;
            if (c < 10)
                out[row * 10 + c] = val;
            else if (c < 22)
                out[OUT_CATE_OFF + row * 12 + (c - 10)] = val;
            else
                out[OUT_POLA_OFF + row * 3 + (c - 22)] = val;
        }
    }
}

// ---------------------------------------------------------------------------
extern "C" void kernel_launch(void* const* d_in, const int* in_sizes, int n_in,
                              void* d_out, int out_size, void* d_ws, size_t ws_size,
                              hipStream_t stream) {
    const float* hidden     = (const float*)d_in[0];
    const int*   rel_pairs  = (const int*)  d_in[1];
    const int*   rel_masks  = (const int*)  d_in[2];
    const int*   rel_sizes  = (const int*)  d_in[3];
    const float* span_reps  = (const float*)d_in[4];
    const float* size_embs  = (const float*)d_in[5];
    const float* size_table = (const float*)d_in[6];
    const float* Wt = (const float*)d_in[7];
    const float* bt = (const float*)d_in[8];
    const float* Wc = (const float*)d_in[9];
    const float* bc = (const float*)d_in[10];
    const float* Wp = (const float*)d_in[11];
    const float* bp = (const float*)d_in[12];
    float* out = (float*)d_out;

    // workspace layout
    float* feat = (float*)d_ws;                          // 512*KPAD floats
    float* PB   = feat + (size_t)B_ * R_ * KPAD;         // 2*KG*64 floats
    float* bias = PB + (size_t)2 * KG * 64;              // 32 floats

    rel_build_features<<<B_ * (R_ / TILE_R), 256, 0, stream>>>(
        hidden, rel_pairs, rel_masks, rel_sizes, span_reps, size_embs,
        size_table, feat);

    rel_pack_weights<<<(2 * KG * 64 + 255) / 256, 256, 0, stream>>>(
        Wt, bt, Wc, bc, Wp, bp, PB, bias);

    rel_gemm_wmma<<<(B_ * R_) / 16 / 4, 128, 0, stream>>>(feat, PB, bias, out);
}